// AbundanceWeightedPooling_58317065945694
// MI455X (gfx1250) — compile-verified
//
#include <hip/hip_runtime.h>
#include <hip/hip_bf16.h>

// ---------- Problem constants ----------
#define N_OTUS    8192
#define BATCH     64
#define SEQ_DIM   256
#define EMBED_DIM 256
#define N_HEADS   4
#define HEAD_DIM  64
#define LN_EPS    1e-5f
#define ATTN_PLANE (BATCH * N_OTUS)     // 524288 floats per head plane
#define KSPLIT    8

typedef __attribute__((ext_vector_type(2))) float v2f;
typedef __attribute__((ext_vector_type(8))) float v8f;

// D = A(16x4 f32) * B(4x16 f32) + C(16x16 f32), wave32 WMMA
__device__ __forceinline__ v8f wmma_f32(v2f a, v2f b, v8f c) {
    return __builtin_amdgcn_wmma_f32_16x16x4_f32(
        /*neg_a=*/false, a, /*neg_b=*/false, b,
        /*c_mod=*/(short)0, c, /*reuse_a=*/false, /*reuse_b=*/false);
}

// ------------------------------------------------------------------
// Kernel 1: values[8192,256] = seq @ value_W.T + value_b   (WMMA f32)
// grid.x = 512 M-tiles, block = 256 (8 waves), wave w -> N-tiles {2w, 2w+1}
// ------------------------------------------------------------------
__global__ void k_values(const float* __restrict__ seq,
                         const float* __restrict__ W,
                         const float* __restrict__ bias,
                         float* __restrict__ out) {
    const int wave = threadIdx.x >> 5;
    const int lane = threadIdx.x & 31;
    const int half = lane >> 4;      // 0: K,K+1   1: K+2,K+3
    const int lr   = lane & 15;      // M (for A) / N (for B,C)
    const int m0   = blockIdx.x * 16;

    #pragma unroll
    for (int t = 0; t < 2; ++t) {
        const int n0 = (wave * 2 + t) * 16;
        v8f c = {};
        const float* arow = seq + (m0 + lr) * SEQ_DIM;
        const float* brow = W   + (n0 + lr) * SEQ_DIM;   // B[k][n] = W[n][k]
        for (int kb = 0; kb < SEQ_DIM; kb += 4) {
            const int ka = kb + half * 2;
            float2 af = *(const float2*)(arow + ka);
            float2 bf = *(const float2*)(brow + ka);
            v2f a; a.x = af.x; a.y = af.y;
            v2f b; b.x = bf.x; b.y = bf.y;
            c = wmma_f32(a, b, c);
        }
        const float bn = bias[n0 + lr];
        #pragma unroll
        for (int r = 0; r < 8; ++r) {
            const int M = r + half * 8;
            out[(m0 + M) * EMBED_DIM + n0 + lr] = c[r] + bn;
        }
    }
}

// ------------------------------------------------------------------
// Kernel 2: base_scores[8192,4] = seq @ score_W.T + score_b
// one wave per row; block = 8 waves; grid = 1024
// ------------------------------------------------------------------
__global__ void k_scores(const float* __restrict__ seq,
                         const float* __restrict__ sW,
                         const float* __restrict__ sb,
                         float* __restrict__ out) {
    const int warp = threadIdx.x >> 5;
    const int lane = threadIdx.x & 31;
    const int m = blockIdx.x * 8 + warp;
    float acc[N_HEADS] = {0.f, 0.f, 0.f, 0.f};
    const float* row = seq + m * SEQ_DIM;
    for (int k = lane; k < SEQ_DIM; k += 32) {
        const float s = row[k];
        #pragma unroll
        for (int h = 0; h < N_HEADS; ++h) acc[h] = fmaf(s, sW[h * SEQ_DIM + k], acc[h]);
    }
    #pragma unroll
    for (int off = 16; off > 0; off >>= 1) {
        #pragma unroll
        for (int h = 0; h < N_HEADS; ++h) acc[h] += __shfl_down(acc[h], off, 32);
    }
    if (lane == 0) {
        #pragma unroll
        for (int h = 0; h < N_HEADS; ++h) out[m * N_HEADS + h] = acc[h] + sb[h];
    }
}

// ------------------------------------------------------------------
// Kernel 3: masked softmax over n for each (b,h); logits cached in LDS
// grid = (64, 4), block = 256, 32 KB static LDS
// attn weights written PLANAR: attnw[h][b][n]
// ------------------------------------------------------------------
__global__ void k_softmax(const float* __restrict__ scores,
                          const float* __restrict__ clr,
                          const unsigned char* __restrict__ mask,
                          const float* __restrict__ gate_W,
                          const float* __restrict__ gate_b,
                          float* __restrict__ attnw) {
    __shared__ float sl[N_OTUS];
    __shared__ float red[8];
    const int b = blockIdx.x, h = blockIdx.y;
    const int t = threadIdx.x, lane = t & 31, warp = t >> 5;
    const float gw = gate_W[h];       // gate_W shape (4,1)
    const float gb = gate_b[h];
    const float NEG = -3.0e38f;

    float pmax = NEG;
    for (int i = 0; i < N_OTUS / 256; ++i) {
        const int n = i * 256 + t;
        float lg;
        if (mask[b * N_OTUS + n]) {
            lg = NEG;
        } else {
            const float sig = tanhf(fmaf(clr[b * N_OTUS + n], gw, gb));
            lg = scores[n * N_HEADS + h] * sig;
        }
        sl[n] = lg;
        pmax = fmaxf(pmax, lg);
    }
    #pragma unroll
    for (int off = 16; off > 0; off >>= 1) pmax = fmaxf(pmax, __shfl_down(pmax, off, 32));
    if (lane == 0) red[warp] = pmax;
    __syncthreads();
    if (t == 0) { float m = red[0]; for (int i = 1; i < 8; ++i) m = fmaxf(m, red[i]); red[0] = m; }
    __syncthreads();
    const float mx = red[0];
    __syncthreads();

    float psum = 0.f;
    for (int i = 0; i < N_OTUS / 256; ++i) {
        const int n = i * 256 + t;
        const float lg = sl[n];
        const float e = (lg <= -1.0e38f) ? 0.f : __expf(lg - mx);
        sl[n] = e;
        psum += e;
    }
    #pragma unroll
    for (int off = 16; off > 0; off >>= 1) psum += __shfl_down(psum, off, 32);
    if (lane == 0) red[warp] = psum;
    __syncthreads();
    if (t == 0) { float s = 0.f; for (int i = 0; i < 8; ++i) s += red[i]; red[0] = s; }
    __syncthreads();
    const float inv = 1.0f / red[0];

    float* plane = attnw + (size_t)h * ATTN_PLANE + (size_t)b * N_OTUS;
    for (int i = 0; i < N_OTUS / 256; ++i) {
        const int n = i * 256 + t;
        plane[n] = sl[n] * inv;
    }
}

// ------------------------------------------------------------------
// Kernel 4: avg_attn[b,n] = mean over head planes (coalesced streams)
// ------------------------------------------------------------------
__global__ void k_avg(const float* __restrict__ attnw, float* __restrict__ avg) {
    const int i = blockIdx.x * 256 + threadIdx.x;   // 0 .. B*N_OTUS-1
    avg[i] = (attnw[i] + attnw[ATTN_PLANE + i] +
              attnw[2 * ATTN_PLANE + i] + attnw[3 * ATTN_PLANE + i]) * 0.25f;
}

// ------------------------------------------------------------------
// Kernel 5: zero the pooled accumulator (re-init every launch)
// ------------------------------------------------------------------
__global__ void k_zero(float* __restrict__ p) {
    p[blockIdx.x * 256 + threadIdx.x] = 0.f;
}

// ------------------------------------------------------------------
// Kernel 6: pooled[b, h*64+d] += sum_n attn[h][b][n] * values[n, h*64+d]
// per head: 64x8192 @ 8192x64 via WMMA f32, K split 8x + atomic reduce
// grid = (4 heads, 8 k-chunks), block = 512 (16 waves, one 16x16 tile each)
// A rows contiguous in k (planar attn) -> single b64 load per lane per step
// ------------------------------------------------------------------
__global__ void __launch_bounds__(512) k_pool(const float* __restrict__ attnw,
                                              const float* __restrict__ values,
                                              float* __restrict__ pooled) {
    const int wave = threadIdx.x >> 5;
    const int lane = threadIdx.x & 31;
    const int half = lane >> 4;
    const int lr   = lane & 15;
    const int h  = blockIdx.x;
    const int kc = blockIdx.y;
    const int mt = wave >> 2;          // 0..3  (B tile)
    const int nt = wave & 3;           // 0..3  (head-dim tile)

    v8f c = {};
    const int k0 = kc * (N_OTUS / KSPLIT);
    const int k1 = k0 + (N_OTUS / KSPLIT);
    const int m = mt * 16 + lr;
    const int n = nt * 16 + lr;
    const float* arow = attnw + (size_t)h * ATTN_PLANE + (size_t)m * N_OTUS; // contiguous k
    const float* bcol = values + h * HEAD_DIM + n;                           // stride 256 in k
    for (int kb = k0; kb < k1; kb += 4) {
        const int ka = kb + half * 2;
        // prefetch the strided values stream 64 k-steps ahead (speculative)
        __builtin_prefetch(bcol + (size_t)(ka + 64) * EMBED_DIM, 0, 1);
        float2 af = *(const float2*)(arow + ka);
        v2f a; a.x = af.x; a.y = af.y;
        v2f b;
        b.x = bcol[(size_t)ka * EMBED_DIM];
        b.y = bcol[(size_t)(ka + 1) * EMBED_DIM];
        c = wmma_f32(a, b, c);
    }
    #pragma unroll
    for (int r = 0; r < 8; ++r) {
        const int M = mt * 16 + r + half * 8;
        atomicAdd(&pooled[M * EMBED_DIM + h * HEAD_DIM + nt * 16 + lr], c[r]);
    }
}

// ------------------------------------------------------------------
// Kernel 7: h = gelu(pooled @ out_W.T + out_b); LayerNorm -> output
// one block per batch row; 256 threads (one per output feature)
// ------------------------------------------------------------------
__global__ void k_head(const float* __restrict__ pooled,
                       const float* __restrict__ oW,
                       const float* __restrict__ ob,
                       const float* __restrict__ gamma,
                       const float* __restrict__ beta,
                       float* __restrict__ out) {
    __shared__ float prow[EMBED_DIM];
    __shared__ float r1[8], r2[8];
    __shared__ float stat[2];
    const int b = blockIdx.x, d = threadIdx.x, lane = d & 31, warp = d >> 5;
    prow[d] = pooled[b * EMBED_DIM + d];
    __syncthreads();

    float acc = ob[d];
    const float* wrow = oW + d * EMBED_DIM;
    #pragma unroll 8
    for (int k = 0; k < EMBED_DIM; ++k) acc = fmaf(prow[k], wrow[k], acc);

    // exact GELU
    const float g = 0.5f * acc * (1.0f + erff(acc * 0.70710678118654752440f));

    float v1 = g, v2 = g * g;
    #pragma unroll
    for (int off = 16; off > 0; off >>= 1) {
        v1 += __shfl_down(v1, off, 32);
        v2 += __shfl_down(v2, off, 32);
    }
    if (lane == 0) { r1[warp] = v1; r2[warp] = v2; }
    __syncthreads();
    if (d == 0) {
        float a = 0.f, q = 0.f;
        for (int i = 0; i < 8; ++i) { a += r1[i]; q += r2[i]; }
        const float mu = a * (1.0f / EMBED_DIM);
        stat[0] = mu;
        stat[1] = q * (1.0f / EMBED_DIM) - mu * mu;
    }
    __syncthreads();
    const float mu = stat[0], var = stat[1];
    out[b * EMBED_DIM + d] = (g - mu) * rsqrtf(var + LN_EPS) * gamma[d] + beta[d];
}

// ------------------------------------------------------------------
extern "C" void kernel_launch(void* const* d_in, const int* in_sizes, int n_in,
                              void* d_out, int out_size, void* d_ws, size_t ws_size,
                              hipStream_t stream) {
    const float*         seq     = (const float*)d_in[0];          // (8192,256)
    const float*         clr     = (const float*)d_in[1];          // (64,8192)
    const unsigned char* mask    = (const unsigned char*)d_in[2];  // (64,8192) bool
    const float*         score_W = (const float*)d_in[3];          // (4,256)
    const float*         score_b = (const float*)d_in[4];          // (4,)
    const float*         gate_W  = (const float*)d_in[5];          // (4,1)
    const float*         gate_b  = (const float*)d_in[6];          // (4,)
    const float*         value_W = (const float*)d_in[7];          // (256,256)
    const float*         value_b = (const float*)d_in[8];          // (256,)
    const float*         out_W   = (const float*)d_in[9];          // (256,256)
    const float*         out_b   = (const float*)d_in[10];         // (256,)
    const float*         ln_g    = (const float*)d_in[11];         // (256,)
    const float*         ln_b    = (const float*)d_in[12];         // (256,)

    // workspace layout (floats)
    float* ws_values = (float*)d_ws;                               // 8192*256
    float* ws_scores = ws_values + (size_t)N_OTUS * EMBED_DIM;     // 8192*4
    float* ws_attn   = ws_scores + (size_t)N_OTUS * N_HEADS;       // 4 planes of 64*8192
    float* ws_pooled = ws_attn + (size_t)N_HEADS * ATTN_PLANE;     // 64*256

    float* out_main = (float*)d_out;                  // (64,256)
    float* out_avg  = out_main + BATCH * EMBED_DIM;   // (64,8192)

    k_values <<<dim3(N_OTUS / 16), dim3(256), 0, stream>>>(seq, value_W, value_b, ws_values);
    k_scores <<<dim3(N_OTUS / 8),  dim3(256), 0, stream>>>(seq, score_W, score_b, ws_scores);
    k_softmax<<<dim3(BATCH, N_HEADS), dim3(256), 0, stream>>>(ws_scores, clr, mask, gate_W, gate_b, ws_attn);
    k_avg    <<<dim3(BATCH * N_OTUS / 256), dim3(256), 0, stream>>>(ws_attn, out_avg);
    k_zero   <<<dim3(BATCH), dim3(256), 0, stream>>>(ws_pooled);
    k_pool   <<<dim3(N_HEADS, KSPLIT), dim3(512), 0, stream>>>(ws_attn, ws_values, ws_pooled);
    k_head   <<<dim3(BATCH), dim3(256), 0, stream>>>(ws_pooled, out_W, out_b, ln_g, ln_b, out_main);
}